// MMDLoss_40458591928582
// MI455X (gfx1250) — compile-verified
//
#include <hip/hip_runtime.h>
#include <hip/hip_bf16.h>

// MMD loss, multi-bandwidth RBF, via f32 WMMA Gram matrices on gfx1250.
//
// Roofline: ~137 GFLOP f32 matrix work (triangular savings on the symmetric
// terms), inputs (32MB) resident in the 192MB L2 -> bound by the
// V_WMMA_F32_16X16X4_F32 pipe. f32 (not bf16) because the answer is
// dominated by the Kxx/Kyy diagonal where d = ||x||^2+||x||^2-2x.x must
// cancel to ~0 before exp(); low-precision WMMA would wreck that.
//
// Data movement: global->LDS staging uses the CDNA5 async-to-LDS path
// (GLOBAL_LOAD_ASYNC_TO_LDS_B128, ASYNCcnt) with double buffering, so the
// DMA of chunk i+1 overlaps the 64 WMMAs of chunk i. Async loads from one
// wave complete in order, so s_wait_asynccnt<=8 after issuing 8 new
// transfers guarantees the previous chunk is resident.

typedef __attribute__((ext_vector_type(2))) float v2f;
typedef __attribute__((ext_vector_type(8))) float v8f;
typedef int v4i __attribute__((vector_size(16)));
typedef v4i __attribute__((address_space(1)))* gv4p;   // global (device) v4i*
typedef v4i __attribute__((address_space(3)))* lv4p;   // LDS v4i*

#define NROWS 8192
#define KDIM  512
#define BT    128               // macro tile (rows == cols)
#define KC    32                // K chunk staged in LDS
#define KPAD  36                // padded row stride (floats): 16B-aligned rows,
                                // conflict-free ds_load_b64 bank walk
#define NCHUNK (KDIM / KC)      // 16
#define TSIDE (NROWS / BT)      // 64 tile blocks per side

#if __has_builtin(__builtin_amdgcn_global_load_async_to_lds_b128)
#define ASYNC_COPY16(gsrc, ldst)                                               \
    __builtin_amdgcn_global_load_async_to_lds_b128(                            \
        (gv4p)(void*)(gsrc), (lv4p)(void*)(ldst), 0, 0)
#else
// CDNA5 ISA: GLOBAL_LOAD_ASYNC_TO_LDS_B128 vdst(lds), vaddr64, off
#define ASYNC_COPY16(gsrc, ldst)                                               \
    do {                                                                       \
        unsigned _l = (unsigned)(size_t)(ldst);                                \
        unsigned long long _g = (unsigned long long)(size_t)(gsrc);            \
        asm volatile("global_load_async_to_lds_b128 %0, %1, off"               \
                     :: "v"(_l), "v"(_g) : "memory");                          \
    } while (0)
#endif

#if __has_builtin(__builtin_amdgcn_s_wait_asynccnt)
#define WAIT_ASYNC(n) __builtin_amdgcn_s_wait_asynccnt(n)
#else
#define WAIT_ASYNC(n) asm volatile("s_wait_asynccnt %0" ::"n"(n) : "memory")
#endif

__global__ void mmd_zero_out(float* out) { out[0] = 0.0f; }

// One wave per row: ||row||^2 for x rows [0,8192) and y rows [8192,16384).
__global__ __launch_bounds__(256) void mmd_row_norms(
    const float* __restrict__ x, const float* __restrict__ y,
    float* __restrict__ norms)
{
    const int wave = threadIdx.x >> 5;
    const int lane = threadIdx.x & 31;
    const int row  = blockIdx.x * 8 + wave;          // 2048 blocks * 8 = 16384
    const float* src = (row < NROWS) ? (x + (size_t)row * KDIM)
                                     : (y + (size_t)(row - NROWS) * KDIM);
    const float4* p = (const float4*)(src + lane * 16);  // 16 floats per lane
    float s = 0.0f;
#pragma unroll
    for (int i = 0; i < 4; ++i) {
        float4 v = p[i];
        s += v.x * v.x + v.y * v.y + v.z * v.z + v.w * v.w;
    }
#pragma unroll
    for (int off = 16; off >= 1; off >>= 1) s += __shfl_xor(s, off, 32);
    if (lane == 0) norms[row] = s;
}

// Sum over a 128x128 block of sum_bw exp(-0.5*d/bw), d from the Gram tile
// D = A * B^T built with V_WMMA_F32_16X16X4_F32; atomically accumulates
// coef * blockWeight * blockSum into *out.
// tri==1: triangular block mapping (symmetric term), off-diagonal blocks
// weighted x2, exact-diagonal d forced to 0.
__global__ __launch_bounds__(256) void mmd_term(
    const float* __restrict__ A, const float* __restrict__ B,
    const float* __restrict__ An, const float* __restrict__ Bn,
    float* __restrict__ out, int tri, float coef0)
{
    __shared__ float As[2][BT * KPAD];   // double-buffered: 2 x 18KB
    __shared__ float Bs[2][BT * KPAD];   // double-buffered: 2 x 18KB

    // Block -> (bi, bj) tile coordinates.
    int bi, bj;
    if (tri) {
        int f = blockIdx.x;
        bi = 0;
        while (f >= TSIDE - bi) { f -= TSIDE - bi; ++bi; }   // <=64 scalar iters
        bj = bi + f;                                         // bj >= bi
    } else {
        bi = blockIdx.x / TSIDE;
        bj = blockIdx.x % TSIDE;
    }
    const int aRow0 = bi * BT;
    const int bRow0 = bj * BT;

    const int t    = threadIdx.x;
    const int lane = t & 31;
    const int w    = t >> 5;          // wave 0..7
    const int lr   = lane & 15;       // row/col within 16
    const int kh   = (lane >> 4) * 2; // K sub-offset 0 or 2 (A/B operand layout)
    const int rBase = (w & 3) * 32;   // 4 row groups of 32
    const int cBase = (w >> 2) * 64;  // 2 col groups of 64

    v8f acc[2][4];
#pragma unroll
    for (int rt = 0; rt < 2; ++rt)
#pragma unroll
        for (int ct = 0; ct < 4; ++ct)
            acc[rt][ct] = (v8f){0.f, 0.f, 0.f, 0.f, 0.f, 0.f, 0.f, 0.f};

    // Per-thread staging slice: 4 rows 32 apart, one 16B chunk each -> 8 async
    // transfers per thread per chunk (4 for A, 4 for B).
    const int lrow = t >> 3;          // 0..31
    const int c4   = (t & 7) * 4;     // 0..28

#define STAGE_CHUNK(buf, kcv)                                                  \
    do {                                                                       \
        _Pragma("unroll")                                                      \
        for (int p = 0; p < 4; ++p) {                                          \
            const int row = p * 32 + lrow;                                     \
            ASYNC_COPY16(&A[(size_t)(aRow0 + row) * KDIM + (kcv) + c4],        \
                         &As[buf][row * KPAD + c4]);                           \
            ASYNC_COPY16(&B[(size_t)(bRow0 + row) * KDIM + (kcv) + c4],        \
                         &Bs[buf][row * KPAD + c4]);                           \
        }                                                                      \
    } while (0)

    STAGE_CHUNK(0, 0);                 // prologue: chunk 0 -> buffer 0

    for (int ch = 0; ch < NCHUNK; ++ch) {
        const int cur = ch & 1;
        // Overlap: issue chunk ch+1 into the other buffer (safe: all waves
        // finished reading it at the barrier ending iteration ch-1).
        if (ch + 1 < NCHUNK) {
            STAGE_CHUNK(cur ^ 1, (ch + 1) * KC);
            WAIT_ASYNC(8);             // chunk ch resident; 8 new still in flight
        } else {
            WAIT_ASYNC(0);
        }
        __syncthreads();               // whole buffer visible to all waves

        const float* as = As[cur];
        const float* bs = Bs[cur];
#pragma unroll
        for (int ks = 0; ks < KC / 4; ++ks) {
            const int k0 = ks * 4 + kh;
            // Gram trick: A-operand (16x4) and B-operand (4x16) per-lane data
            // are the same function of (lr, kh) over row-major source rows.
            v2f a0 = *(const v2f*)&as[(rBase      + lr) * KPAD + k0];
            v2f a1 = *(const v2f*)&as[(rBase + 16 + lr) * KPAD + k0];
            v2f b0 = *(const v2f*)&bs[(cBase      + lr) * KPAD + k0];
            v2f b1 = *(const v2f*)&bs[(cBase + 16 + lr) * KPAD + k0];
            v2f b2 = *(const v2f*)&bs[(cBase + 32 + lr) * KPAD + k0];
            v2f b3 = *(const v2f*)&bs[(cBase + 48 + lr) * KPAD + k0];
            acc[0][0] = __builtin_amdgcn_wmma_f32_16x16x4_f32(false, a0, false, b0, (short)0, acc[0][0], false, false);
            acc[0][1] = __builtin_amdgcn_wmma_f32_16x16x4_f32(false, a0, false, b1, (short)0, acc[0][1], false, false);
            acc[0][2] = __builtin_amdgcn_wmma_f32_16x16x4_f32(false, a0, false, b2, (short)0, acc[0][2], false, false);
            acc[0][3] = __builtin_amdgcn_wmma_f32_16x16x4_f32(false, a0, false, b3, (short)0, acc[0][3], false, false);
            acc[1][0] = __builtin_amdgcn_wmma_f32_16x16x4_f32(false, a1, false, b0, (short)0, acc[1][0], false, false);
            acc[1][1] = __builtin_amdgcn_wmma_f32_16x16x4_f32(false, a1, false, b1, (short)0, acc[1][1], false, false);
            acc[1][2] = __builtin_amdgcn_wmma_f32_16x16x4_f32(false, a1, false, b2, (short)0, acc[1][2], false, false);
            acc[1][3] = __builtin_amdgcn_wmma_f32_16x16x4_f32(false, a1, false, b3, (short)0, acc[1][3], false, false);
        }
        __syncthreads();               // all waves done reading buf[cur]
    }
#undef STAGE_CHUNK

    // Epilogue: d_ij = ||a_i||^2 + ||b_j||^2 - 2*G_ij, multi-bw RBF, reduce.
    const float coef = coef0 * ((tri && bi != bj) ? 2.0f : 1.0f);
    const int hi = lane >> 4;         // C/D layout: VGPR e -> M = e + 8*hi
    float local = 0.0f;
#pragma unroll
    for (int rt = 0; rt < 2; ++rt) {
        const int iBase = aRow0 + rBase + rt * 16 + hi * 8;
        const float* xn = An + iBase;
#pragma unroll
        for (int ct = 0; ct < 4; ++ct) {
            const int j  = bRow0 + cBase + ct * 16 + lr;
            const float yn = Bn[j];
#pragma unroll
            for (int e = 0; e < 8; ++e) {
                float d = xn[e] + yn - 2.0f * acc[rt][ct][e];
                d = fmaxf(d, 0.0f);
                if (tri && (iBase + e == j)) d = 0.0f;   // exact diagonal
                local += __expf(-0.5f * d)               // bw = 1
                       + __expf(-0.25f * d)              // bw = 2
                       + __expf(-0.16666667f * d);       // bw = 3
            }
        }
    }
    local *= coef;
#pragma unroll
    for (int off = 16; off >= 1; off >>= 1) local += __shfl_xor(local, off, 32);
    if (lane == 0) atomicAdd(out, local);
}

extern "C" void kernel_launch(void* const* d_in, const int* in_sizes, int n_in,
                              void* d_out, int out_size, void* d_ws, size_t ws_size,
                              hipStream_t stream) {
    const float* x = (const float*)d_in[0];
    const float* y = (const float*)d_in[1];
    float* out   = (float*)d_out;
    float* norms = (float*)d_ws;          // 16384 floats (x norms, then y norms)

    mmd_zero_out<<<1, 1, 0, stream>>>(out);
    mmd_row_norms<<<(2 * NROWS) / 8, 256, 0, stream>>>(x, y, norms);

    const float invN2 = 1.0f / ((float)NROWS * (float)NROWS);
    const int triBlocks = TSIDE * (TSIDE + 1) / 2;

    // +mean(Kxx) +mean(Kyy) -2*mean(Kxy)
    mmd_term<<<triBlocks, 256, 0, stream>>>(x, x, norms, norms, out, 1, invN2);
    mmd_term<<<triBlocks, 256, 0, stream>>>(y, y, norms + NROWS, norms + NROWS, out, 1, invN2);
    mmd_term<<<TSIDE * TSIDE, 256, 0, stream>>>(x, y, norms, norms + NROWS, out, 0, -2.0f * invN2);
}